// LowRankLoss_66838281060761
// MI455X (gfx1250) — compile-verified
//
#include <hip/hip_runtime.h>

typedef float v2f __attribute__((ext_vector_type(2)));
typedef float v4f __attribute__((ext_vector_type(4)));
typedef float v8f __attribute__((ext_vector_type(8)));

#define Nn 128
#define Cc 256
#define Hh 32
#define Ww 64
#define HW (Hh * Ww)      // 2048
#define MPITCH 65         // 32x64 tile padded: breaks stride-64 LDS bank conflicts
#define GPITCH 33         // 32x32 Gram padded

// One block per (n, tensor). 256 threads = 8 wave32.
// Streams 2 MB of feature data (channel mean), normalizes, builds the 32x32
// Gram matrix with V_WMMA_F32_16X16X4_F32, then counts positive pivots.
__global__ __launch_bounds__(256)
void lowrank_rank_kernel(const float* __restrict__ raw,
                         const float* __restrict__ rect,
                         float* __restrict__ ranks) {
    __shared__ float Ms[Hh * MPITCH];   // normalized channel-mean matrix, 32x65
    __shared__ float Gs[Hh * GPITCH];   // Gram matrix, 32x33
    __shared__ float ssq;

    const int n   = blockIdx.x;
    const int t   = blockIdx.y;
    const int tid = threadIdx.x;

    const float* __restrict__ feat = (t == 0) ? raw : rect;
    const v4f*  __restrict__ base4 =
        (const v4f*)(feat + (size_t)n * Cc * HW);

    // ---------------- Phase 1: channel mean (memory-bound streaming) -------
    // 2048 floats = 512 float4; thread owns float4 slots {tid, tid+256}.
    v4f a0 = {0.f, 0.f, 0.f, 0.f};
    v4f a1 = {0.f, 0.f, 0.f, 0.f};
    for (int c = 0; c < Cc; ++c) {
        const v4f* p = base4 + (size_t)c * (HW / 4);
        a0 += __builtin_nontemporal_load(p + tid);          // global_load_b128, NT
        a1 += __builtin_nontemporal_load(p + tid + 256);
    }

    if (tid == 0) ssq = 0.f;
    __syncthreads();

    const float invC = 1.f / (float)Cc;
    a0 *= invC;
    a1 *= invC;

    float local = 0.f;
    #pragma unroll
    for (int e = 0; e < 4; ++e) {
        int p0 = 4 * tid + e;
        int p1 = 4 * (tid + 256) + e;
        Ms[(p0 >> 6) * MPITCH + (p0 & 63)] = a0[e];
        Ms[(p1 >> 6) * MPITCH + (p1 & 63)] = a1[e];
        local += a0[e] * a0[e] + a1[e] * a1[e];
    }
    atomicAdd(&ssq, local);             // ds_add_f32
    __syncthreads();

    // Frobenius normalization (scales singular values; keeps rank semantics)
    const float scale = rsqrtf(ssq);
    #pragma unroll
    for (int e = 0; e < 4; ++e) {
        int p0 = 4 * tid + e;
        int p1 = 4 * (tid + 256) + e;
        Ms[(p0 >> 6) * MPITCH + (p0 & 63)] = a0[e] * scale;
        Ms[(p1 >> 6) * MPITCH + (p1 & 63)] = a1[e] * scale;
    }
    __syncthreads();

    // ---------------- Phase 2: G = M * M^T via WMMA f32 16x16x4 ------------
    const int wv   = tid >> 5;
    const int lane = tid & 31;
    if (wv < 4) {                       // uniform per wave -> EXEC all ones
        const int rowT = (wv >> 1) * 16;
        const int colT = (wv & 1)  * 16;
        const int half = lane >> 4;     // 0: K={k,k+1}; 1: K={k+2,k+3}
        const int l    = lane & 15;

        v8f acc = {};
        for (int k = 0; k < Ww; k += 4) {
            const int ka = k + 2 * half;
            v2f a, b;
            // A tile: rows rowT..rowT+15 of M          (16x4, f32 layout)
            a.x = Ms[(rowT + l) * MPITCH + ka];
            a.y = Ms[(rowT + l) * MPITCH + ka + 1];
            // B tile: M^T -> columns = rows colT..colT+15 of M (4x16)
            b.x = Ms[(colT + l) * MPITCH + ka];
            b.y = Ms[(colT + l) * MPITCH + ka + 1];
            acc = __builtin_amdgcn_wmma_f32_16x16x4_f32(
                false, a, false, b, (short)0, acc, false, false);
        }
        // D layout: VGPR r -> M = r (lanes 0-15) / r+8 (lanes 16-31), N = l
        #pragma unroll
        for (int r = 0; r < 8; ++r) {
            int row = rowT + (half ? 8 : 0) + r;
            Gs[row * GPITCH + colT + l] = acc[r];
        }
    }
    __syncthreads();

    // ---------------- Phase 3: rank = #positive pivots (wave 0) ------------
    // G is PSD -> unpivoted symmetric elimination is stable; TOL == 0.
    if (tid < 32) {
        const int j = tid;              // lane owns column j
        int rank = 0;
        for (int k = 0; k < Hh; ++k) {
            float piv = Gs[k * GPITCH + k];     // broadcast read
            if (piv > 0.f) rank++;
            float invp = (piv != 0.f) ? (1.f / piv) : 0.f;
            float rowk = Gs[k * GPITCH + j];
            for (int i = k + 1; i < Hh; ++i) {
                float fac = Gs[i * GPITCH + k] * invp;  // read before write (in-order DS)
                Gs[i * GPITCH + j] -= fac * rowk;
            }
        }
        if (j == 0) ranks[t * Nn + n] = (float)rank;
    }
}

// Margin ranking loss: mean_n max(0, -(rank1 - rank2))   (y=+1, margin=0)
__global__ __launch_bounds__(128)
void lowrank_loss_kernel(const float* __restrict__ ranks,
                         float* __restrict__ out) {
    __shared__ float red[Nn];
    const int n = threadIdx.x;
    float r1 = ranks[n];        // raw_feat ranks
    float r2 = ranks[Nn + n];   // rectified_feat ranks
    red[n] = fmaxf(0.f, -(r1 - r2));
    __syncthreads();
    for (int s = Nn / 2; s > 0; s >>= 1) {
        if (n < s) red[n] += red[n + s];
        __syncthreads();
    }
    if (n == 0) out[0] = red[0] / (float)Nn;
}

extern "C" void kernel_launch(void* const* d_in, const int* in_sizes, int n_in,
                              void* d_out, int out_size, void* d_ws, size_t ws_size,
                              hipStream_t stream) {
    (void)in_sizes; (void)n_in; (void)out_size; (void)ws_size;

    const float* raw   = (const float*)d_in[0];
    const float* rect  = (const float*)d_in[1];
    float*       ranks = (float*)d_ws;          // 2*128 floats, fully rewritten
    float*       out   = (float*)d_out;

    dim3 grid(Nn, 2);
    lowrank_rank_kernel<<<grid, 256, 0, stream>>>(raw, rect, ranks);
    lowrank_loss_kernel<<<1, Nn, 0, stream>>>(ranks, out);
}